// MAGG_Module_88433376624847
// MI455X (gfx1250) — compile-verified
//
#include <hip/hip_runtime.h>

typedef __attribute__((ext_vector_type(2))) float v2f;
typedef __attribute__((ext_vector_type(8))) float v8f;

#define N_NODES 40000
#define N_EDGES 640000
#define DIM     128
#define KTOT    (2 * DIM)      // concatenated [mean || x] K dimension
#define APAD    (KTOT + 4)     // LDS row stride: 260 floats -> stride%64==4 -> conflict-free

__device__ __forceinline__ void atomic_add_f32(float* p, float v) {
    // relaxed agent-scope float add -> lowers to global_atomic_add_f32 (no return)
    __hip_atomic_fetch_add(p, v, __ATOMIC_RELAXED, __HIP_MEMORY_SCOPE_AGENT);
}

__global__ void zero_kernel(float* __restrict__ p, int n) {
    int i = blockIdx.x * blockDim.x + threadIdx.x;
    if (i < n) p[i] = 0.0f;
}

__global__ void degree_kernel(const int* __restrict__ ei, float* __restrict__ cnt) {
    int e = blockIdx.x * blockDim.x + threadIdx.x;
    if (e < N_EDGES) {
        int dst = ei[N_EDGES + e];
        atomic_add_f32(&cnt[dst], 1.0f);
    }
}

// One wave32 per edge; each lane moves 4 contiguous features (float4 load + 4 hw atomics).
__global__ void scatter_kernel(const int* __restrict__ ei, const float* __restrict__ x,
                               float* __restrict__ agg) {
    int gid  = blockIdx.x * blockDim.x + threadIdx.x;
    int edge = gid >> 5;
    int lane = gid & 31;
    if (edge >= N_EDGES) return;
    int src = ei[edge];
    int dst = ei[N_EDGES + edge];
    const float4 v = *(const float4*)(x + (size_t)src * DIM + lane * 4);
    float* d = agg + (size_t)dst * DIM + lane * 4;
    atomic_add_f32(d + 0, v.x);
    atomic_add_f32(d + 1, v.y);
    atomic_add_f32(d + 2, v.z);
    atomic_add_f32(d + 3, v.w);
}

// Fused SAGEConv GEMM: out[i,:] = act( (agg[i,:]/max(cnt,1)) @ Wl + xin[i,:] @ Wr + bias )
// Block = 16 node rows x full 128 output cols; 8 waves, one 16x16 tile per wave.
// K = 256 via v_wmma_f32_16x16x4_f32 chain (64 WMMAs per wave).
__global__ __launch_bounds__(256) void sage_gemm_kernel(
    const float* __restrict__ agg, const float* __restrict__ cnt,
    const float* __restrict__ xin, const float* __restrict__ Wl,
    const float* __restrict__ Wr,  const float* __restrict__ bias,
    float* __restrict__ out, int relu) {

    __shared__ float sA[16][APAD];   // 16 rows x (mean||x), padded
    __shared__ float sInv[16];

    const int tid     = threadIdx.x;
    const int lane    = tid & 31;
    const int waveId  = tid >> 5;          // 0..7 -> output col tile
    const int rowBase = blockIdx.x * 16;

    if (tid < 16) {
        sInv[tid] = 1.0f / fmaxf(cnt[rowBase + tid], 1.0f);
    }
    __syncthreads();

    // Stage A strip: cols [0,128) = mean (agg * 1/cnt), cols [128,256) = xin. Coalesced.
    for (int idx = tid; idx < 16 * KTOT; idx += 256) {
        int r = idx >> 8;        // / 256
        int c = idx & 255;
        float v;
        if (c < DIM) v = agg[(size_t)(rowBase + r) * DIM + c] * sInv[r];
        else         v = xin[(size_t)(rowBase + r) * DIM + (c - DIM)];
        sA[r][c] = v;
    }
    __syncthreads();

    const int n0     = waveId * 16;
    const int laneN  = lane & 15;          // M for A-frag, N for B-frag
    const int laneHi = lane >> 4;          // 0: K0/K1 half, 1: K2/K3 half

    v8f acc = {};
    #pragma unroll 4
    for (int k = 0; k < KTOT; k += 4) {
        // A fragment (16x4 f32): lanes0-15 -> K = k+0,k+1 ; lanes16-31 -> K = k+2,k+3
        int kk = k + 2 * laneHi;
        v2f a;
        a.x = sA[laneN][kk];
        a.y = sA[laneN][kk + 1];

        // B fragment (4x16 f32) from combined [Wl ; Wr]
        const float* W = (k < DIM) ? Wl : Wr;
        int kl = (k < DIM) ? k : (k - DIM);
        int kr = kl + 2 * laneHi;
        v2f b;
        b.x = W[(size_t)kr       * DIM + n0 + laneN];
        b.y = W[(size_t)(kr + 1) * DIM + n0 + laneN];

        // D = A*B + C  (8 args: neg_a, A, neg_b, B, c_mod, C, reuse_a, reuse_b)
        acc = __builtin_amdgcn_wmma_f32_16x16x4_f32(
            false, a, false, b, (short)0, acc, false, false);
    }

    // C/D layout: VGPR r -> M = r + 8*laneHi ; N = laneN
    const float bv = bias[n0 + laneN];
    #pragma unroll
    for (int r = 0; r < 8; ++r) {
        int m = r + 8 * laneHi;
        float v = acc[r] + bv;
        if (relu) v = fmaxf(v, 0.0f);
        out[(size_t)(rowBase + m) * DIM + (n0 + laneN)] = v;
    }
}

extern "C" void kernel_launch(void* const* d_in, const int* in_sizes, int n_in,
                              void* d_out, int out_size, void* d_ws, size_t ws_size,
                              hipStream_t stream) {
    const float* x   = (const float*)d_in[0];
    const int*   ei  = (const int*)  d_in[1];
    const float* W1l = (const float*)d_in[2];
    const float* b1  = (const float*)d_in[3];
    const float* W1r = (const float*)d_in[4];
    const float* W2l = (const float*)d_in[5];
    const float* b2  = (const float*)d_in[6];
    const float* W2r = (const float*)d_in[7];
    float* out = (float*)d_out;

    float* agg = (float*)d_ws;                         // N*128 f32
    float* h   = agg + (size_t)N_NODES * DIM;          // N*128 f32
    float* cnt = h   + (size_t)N_NODES * DIM;          // N f32

    const int feat_elems = N_NODES * DIM;

    // ---- Layer 1 ----
    zero_kernel<<<(feat_elems + 255) / 256, 256, 0, stream>>>(agg, feat_elems);
    zero_kernel<<<(N_NODES + 255) / 256, 256, 0, stream>>>(cnt, N_NODES);
    degree_kernel<<<(N_EDGES + 255) / 256, 256, 0, stream>>>(ei, cnt);
    scatter_kernel<<<(N_EDGES * 32) / 256, 256, 0, stream>>>(ei, x, agg);
    sage_gemm_kernel<<<N_NODES / 16, 256, 0, stream>>>(agg, cnt, x, W1l, W1r, b1, h, /*relu=*/1);

    // ---- Layer 2 (same graph -> cnt reused) ----
    zero_kernel<<<(feat_elems + 255) / 256, 256, 0, stream>>>(agg, feat_elems);
    scatter_kernel<<<(N_EDGES * 32) / 256, 256, 0, stream>>>(ei, h, agg);
    sage_gemm_kernel<<<N_NODES / 16, 256, 0, stream>>>(agg, cnt, h, W2l, W2r, b2, out, /*relu=*/0);
}